// VolumeRenderer_386547057304
// MI455X (gfx1250) — compile-verified
//
#include <hip/hip_runtime.h>
#include <stdint.h>

#define G 128
#define TY 8
#define THRESH 0.5f

// exposed face = center occupied AND neighbor empty
__device__ __forceinline__ float face(bool cv, float nb) {
    return (cv && !(nb > THRESH)) ? 1.0f : 0.0f;
}

__global__ __launch_bounds__(256) void cubify_faces_kernel(
    const float* __restrict__ dens, float* __restrict__ out)
{
    // ---- LDS staging buffers ----
    __shared__ float sC  [(TY + 2) * G];   // center plane rows y0-1 .. y0+8
    __shared__ float sXm [ TY      * G];   // plane x-1, rows y0 .. y0+7
    __shared__ float sXp [ TY      * G];   // plane x+1, rows y0 .. y0+7
    __shared__ float sOut[ TY * G * 6 ];   // per-warp 3072B output slabs (24KB)

    const int tid  = threadIdx.x;
    const int lane = tid & 31;
    const int warp = tid >> 5;                 // 0..7
    const int bid  = blockIdx.x;
    const int yblk = bid & 15;                 // G/TY = 16
    const int x    = (bid >> 4) & (G - 1);
    const int b    = bid >> 11;                // / (16*128)
    const int y0   = yblk * TY;

    const int planeBase = (b * G + x) * G;     // flat row index of (b,x,y=0)

    // ---------------- stage 26 input rows into LDS via async DMA ----------------
    // r 0..9   : center plane, y = y0-1+r
    // r 10..17 : x-1 plane,    y = y0+(r-10)
    // r 18..25 : x+1 plane,    y = y0+(r-18)
    // Each lane moves one float4 per row (128 floats / 32 lanes).
    #pragma unroll
    for (int k = 0; k < 4; ++k) {
        const int r = warp + 8 * k;            // wave-uniform
        if (r < 26) {
            float* ldst; int y, xx;
            if (r < 10)      { ldst = &sC [ r       * G]; y = y0 - 1 + r;    xx = x;     }
            else if (r < 18) { ldst = &sXm[(r - 10) * G]; y = y0 + (r - 10); xx = x - 1; }
            else             { ldst = &sXp[(r - 18) * G]; y = y0 + (r - 18); xx = x + 1; }

            float* dst4 = ldst + lane * 4;
            if ((unsigned)y < (unsigned)G && (unsigned)xx < (unsigned)G) {
                const float* src = dens + (size_t)((b * G + xx) * G + y) * G;
                uint32_t ldsOff = (uint32_t)(uintptr_t)dst4;   // low 32 bits = LDS offset
                uint32_t gOff   = (uint32_t)(lane * 16);       // per-lane byte offset
                asm volatile(
                    "global_load_async_to_lds_b128 %0, %1, %2"
                    :
                    : "v"(ldsOff), "v"(gOff), "s"(src)
                    : "memory");
            } else {
                // outside the grid: empty voxels
                *(float4*)dst4 = make_float4(0.f, 0.f, 0.f, 0.f);
            }
        }
    }
    asm volatile("s_wait_asynccnt 0x0" ::: "memory");
    __syncthreads();

    // ---------------- compute 4 voxels: (b, x, y0+warp, z0..z0+3) ----------------
    const int r  = warp;
    const int z0 = lane * 4;

    const float4 c  = *(const float4*)&sC [(r + 1) * G + z0];
    const float4 ym = *(const float4*)&sC [ r      * G + z0];
    const float4 yp = *(const float4*)&sC [(r + 2) * G + z0];
    const float4 xm = *(const float4*)&sXm[ r      * G + z0];
    const float4 xp = *(const float4*)&sXp[ r      * G + z0];
    const float  zm = (z0 == 0)     ? 0.f : sC[(r + 1) * G + z0 - 1];
    const float  zp = (z0 == G - 4) ? 0.f : sC[(r + 1) * G + z0 + 4];

    const bool v0 = c.x > THRESH, v1 = c.y > THRESH, v2 = c.z > THRESH, v3 = c.w > THRESH;

    // face order per voxel: [-x, +x, -y, +y, -z, +z]; 24 floats per thread
    float4 W0 = make_float4(face(v0, xm.x), face(v0, xp.x), face(v0, ym.x), face(v0, yp.x));
    float4 W1 = make_float4(face(v0, zm),   face(v0, c.y),  face(v1, xm.y), face(v1, xp.y));
    float4 W2 = make_float4(face(v1, ym.y), face(v1, yp.y), face(v1, c.x),  face(v1, c.z));
    float4 W3 = make_float4(face(v2, xm.z), face(v2, xp.z), face(v2, ym.z), face(v2, yp.z));
    float4 W4 = make_float4(face(v2, c.y),  face(v2, c.w),  face(v3, xm.w), face(v3, xp.w));
    float4 W5 = make_float4(face(v3, ym.w), face(v3, yp.w), face(v3, c.z),  face(v3, zp));

    // ---------------- stage output slab in LDS (global-identical layout) ----------------
    float4* so = (float4*)(sOut + warp * (G * 6) + lane * 24);
    so[0] = W0; so[1] = W1; so[2] = W2; so[3] = W3; so[4] = W4; so[5] = W5;

    // our wave's LDS writes must land before the DMA engine reads them
    asm volatile("s_wait_dscnt 0x0" ::: "memory");

    // ---------------- drain 3072B contiguous per wave via async LDS->global DMA ----------
    // round k: lane i writes bytes [k*512 + i*16, +16) -> every instruction is a
    // fully-dense 512B wave store. NT hint: output is write-once, keep L2 for densities.
    const float* gbase = out + (size_t)(planeBase + y0 + warp) * (G * 6);  // wave-uniform
    const uint32_t ldsBase = (uint32_t)(uintptr_t)(sOut + warp * (G * 6));
    #pragma unroll
    for (int k = 0; k < 6; ++k) {
        uint32_t off  = (uint32_t)(k * 512 + lane * 16);  // global byte offset
        uint32_t lsrc = ldsBase + off;                    // matching LDS byte address
        asm volatile(
            "global_store_async_from_lds_b128 %0, %1, %2 th:TH_STORE_NT"
            :
            : "v"(off), "v"(lsrc), "s"(gbase)
            : "memory");
    }
    // s_endpgm does an implicit wait-idle, but be explicit about outstanding DMA
    asm volatile("s_wait_asynccnt 0x0" ::: "memory");
}

extern "C" void kernel_launch(void* const* d_in, const int* in_sizes, int n_in,
                              void* d_out, int out_size, void* d_ws, size_t ws_size,
                              hipStream_t stream) {
    // inputs: d_in[0] = points [4, 128^3, 3] (unused), d_in[1] = densities [4, 128^3, 1]
    const float* dens = (const float*)d_in[1];
    float* out = (float*)d_out;

    const int B = 4;
    dim3 grid(B * G * (G / TY));   // 4 * 128 * 16 = 8192 blocks
    dim3 block(256);
    cubify_faces_kernel<<<grid, block, 0, stream>>>(dens, out);
}